// Memory_25400436588550
// MI455X (gfx1250) — compile-verified
//
#include <hip/hip_runtime.h>
#include <math.h>

// ---- geometry (fixed by reference) ----
//  query: (32, 256, 64, 64) fp32 ; keys: (512, 256) fp32
//  N = 131072 rows, d = 256, M = 512
#define NROWS 131072
#define DDIM  256
#define MKEYS 512
#define HWSP  4096      // 64*64 spatial per batch

typedef __attribute__((ext_vector_type(2))) float v2f;
typedef __attribute__((ext_vector_type(8))) float v8f;

// ---- CDNA5 async Global->LDS copy (ASYNCcnt-tracked), via inline asm ------
// LDS byte offset = low 32 bits of the generic pointer (the ISA takes its low 32 bits as the LDS address,
// wave-relative; the async copy's LDS address uses the same base).
__device__ __forceinline__ unsigned lds_off(const void* p)
{
    return (unsigned)(size_t)p;
}
__device__ __forceinline__ void async_copy_b32(unsigned ldsByteOff, const float* g)
{
    asm volatile("global_load_async_to_lds_b32 %0, %1, off"
                 :: "v"(ldsByteOff), "v"(g) : "memory");
}
__device__ __forceinline__ void async_copy_b128(unsigned ldsByteOff, const float* g)
{
    asm volatile("global_load_async_to_lds_b128 %0, %1, off"
                 :: "v"(ldsByteOff), "v"(g) : "memory");
}
__device__ __forceinline__ void async_wait0()
{
    asm volatile("s_wait_asynccnt 0x0" ::: "memory");
}

// ---------------------------------------------------------------------------
// 1) per-position inverse L2 norm + write normalized query into updated_query
//    first-half channels (native (B,C,H,W) layout == concat's first d chans)
// ---------------------------------------------------------------------------
__global__ void k_norm(const float* __restrict__ q,
                       float* __restrict__ rnorm,
                       float* __restrict__ uq)
{
    int p = blockIdx.x * 256 + threadIdx.x;        // global spatial index
    int b = p >> 12, s = p & (HWSP - 1);
    const float* base = q + ((size_t)b * DDIM) * HWSP + s;
    float ss = 0.f;
    for (int c = 0; c < DDIM; ++c) {
        float v = base[(size_t)c * HWSP];
        ss += v * v;
    }
    float rn = 1.0f / fmaxf(sqrtf(ss), 1e-12f);
    rnorm[p] = rn;
    float* ob = uq + ((size_t)b * (2 * DDIM)) * HWSP + s;
    for (int c = 0; c < DDIM; ++c)
        ob[(size_t)c * HWSP] = base[(size_t)c * HWSP] * rn;
}

// ---------------------------------------------------------------------------
// 2) GEMM1: score[N,M] = qr @ keys^T   (fp32 WMMA 16x16x4)
//    block = 128 thr (4 waves); block tile = 16 rows x 64 cols
//    A tile staged RAW via async Global->LDS; rnorm folded into A fragments.
// ---------------------------------------------------------------------------
__global__ void k_gemm1(const float* __restrict__ q,
                        const float* __restrict__ keys,
                        const float* __restrict__ rnorm,
                        float* __restrict__ score)
{
    __shared__ float ldsA[16 * 260];               // 16 rows x 256 k, pad 4
    int tid = threadIdx.x;
    int r0 = blockIdx.x * 16;
    int b = r0 >> 12, s0 = r0 & (HWSP - 1);
    const float* qb = q + ((size_t)b * DDIM) * HWSP + s0;
    for (int j = tid; j < 16 * DDIM; j += 128) {
        int m = j & 15, c = j >> 4;                // coalesced along spatial
        async_copy_b32(lds_off(&ldsA[m * 260 + c]), qb + (size_t)c * HWSP + m);
    }
    async_wait0();
    __syncthreads();

    int wave = tid >> 5, lane = tid & 31;
    int col0 = blockIdx.y * 64 + wave * 16;
    int m = lane & 15, hi = lane >> 4;
    float rn = rnorm[r0 + m];                      // scale folded into A frag
    v8f acc = {};
    const float* aRow = &ldsA[m * 260];
    const float* kRow = keys + (size_t)(col0 + m) * DDIM;  // B[k][n]=keys[n][k]
    __builtin_prefetch(kRow, 0, 1);                // global_prefetch (L2-hot keys)
    for (int k0 = 0; k0 < DDIM; k0 += 4) {
        v2f a = *(const v2f*)(aRow + k0 + 2 * hi);
        a.x *= rn; a.y *= rn;
        v2f bb = *(const v2f*)(kRow + k0 + 2 * hi);
        acc = __builtin_amdgcn_wmma_f32_16x16x4_f32(
                  false, a, false, bb, (short)0, acc, false, false);
    }
    for (int j = 0; j < 8; ++j) {
        int row = r0 + j + 8 * hi;                 // C vgpr j: M=j (+8 hi half)
        score[(size_t)row * MKEYS + col0 + m] = acc[j];
    }
}

// ---------------------------------------------------------------------------
// 3) row softmax (axis=1) + top-2 indices; wave per row
// ---------------------------------------------------------------------------
__global__ void k_rowsoftmax(const float* __restrict__ score,
                             float* __restrict__ sm,
                             int* __restrict__ gidx,
                             int* __restrict__ sid2)
{
    int wave = threadIdx.x >> 5, lane = threadIdx.x & 31;
    int i = blockIdx.x * 8 + wave;
    const float* row = score + (size_t)i * MKEYS;
    float vals[16];
    float mx = -3.402823466e38f;
    float t1 = mx, t2 = mx; int i1 = 0, i2 = 0;
    for (int t = 0; t < 16; ++t) {
        int c = lane + 32 * t;
        float v = row[c];
        vals[t] = v;
        mx = fmaxf(mx, v);
        if (v > t1) { t2 = t1; i2 = i1; t1 = v; i1 = c; }
        else if (v > t2) { t2 = v; i2 = c; }
    }
    for (int off = 16; off > 0; off >>= 1) {
        mx = fmaxf(mx, __shfl_xor(mx, off, 32));
        float o1 = __shfl_xor(t1, off, 32);
        int  oi1 = __shfl_xor(i1, off, 32);
        float o2 = __shfl_xor(t2, off, 32);
        int  oi2 = __shfl_xor(i2, off, 32);
        if (o1 > t1) {
            float n2; int ni2;
            if (t1 > o2) { n2 = t1; ni2 = i1; } else { n2 = o2; ni2 = oi2; }
            t1 = o1; i1 = oi1; t2 = n2; i2 = ni2;
        } else if (o1 > t2) { t2 = o1; i2 = oi1; }
    }
    float sum = 0.f, ex[16];
    for (int t = 0; t < 16; ++t) { ex[t] = expf(vals[t] - mx); sum += ex[t]; }
    for (int off = 16; off > 0; off >>= 1) sum += __shfl_xor(sum, off, 32);
    float inv = 1.0f / sum;
    float* orow = sm + (size_t)i * MKEYS;
    for (int t = 0; t < 16; ++t) orow[lane + 32 * t] = ex[t] * inv;
    if (lane == 0) { gidx[i] = i1; sid2[i] = i2; }
}

// ---------------------------------------------------------------------------
// 4) column (axis=0) softmax support: chunked max / sum-of-exp partials
// ---------------------------------------------------------------------------
__global__ void k_colmax_part(const float* __restrict__ score, float* __restrict__ pmax)
{
    int col = blockIdx.x * 256 + threadIdx.x;
    const float* p = score + (size_t)blockIdx.y * 2048 * MKEYS + col;
    float m = -3.402823466e38f;
    for (int r = 0; r < 2048; ++r) m = fmaxf(m, p[(size_t)r * MKEYS]);
    pmax[(size_t)blockIdx.y * MKEYS + col] = m;
}
__global__ void k_colmax_red(const float* __restrict__ pmax, float* __restrict__ colmax)
{
    int col = blockIdx.x * 256 + threadIdx.x;
    float m = -3.402823466e38f;
    for (int j = 0; j < 64; ++j) m = fmaxf(m, pmax[(size_t)j * MKEYS + col]);
    colmax[col] = m;
}
__global__ void k_colsum_part(const float* __restrict__ score,
                              const float* __restrict__ colmax,
                              float* __restrict__ psum)
{
    int col = blockIdx.x * 256 + threadIdx.x;
    float cm = colmax[col];
    const float* p = score + (size_t)blockIdx.y * 2048 * MKEYS + col;
    float s = 0.f;
    for (int r = 0; r < 2048; ++r) s += expf(p[(size_t)r * MKEYS] - cm);
    psum[(size_t)blockIdx.y * MKEYS + col] = s;
}
__global__ void k_colsum_red(const float* __restrict__ psum, float* __restrict__ colsum)
{
    int col = blockIdx.x * 256 + threadIdx.x;
    float s = 0.f;
    for (int j = 0; j < 64; ++j) s += psum[(size_t)j * MKEYS + col];
    colsum[col] = s;
}
__global__ void k_sq_transform(float* __restrict__ score,
                               const float* __restrict__ colmax,
                               const float* __restrict__ colsum)
{
    size_t idx = (size_t)blockIdx.x * 256 + threadIdx.x;
    int col = (int)(idx & (MKEYS - 1));
    score[idx] = expf(score[idx] - colmax[col]) / colsum[col];
}

// ---------------------------------------------------------------------------
// 5) per-row: gathering loss, triplet loss, weighted segment-sum into qupd.
//    Runs on RAW scores (before in-place transform). w = exp(raw - colmax).
//    Query tile staged via async Global->LDS.
// ---------------------------------------------------------------------------
__global__ void k_rowops(const float* __restrict__ q,
                         const float* __restrict__ keys,
                         const float* __restrict__ rnorm,
                         const float* __restrict__ rawscore,
                         const float* __restrict__ colmax,
                         const int* __restrict__ gidx,
                         const int* __restrict__ sid2,
                         float* __restrict__ qupd,
                         float* __restrict__ accum)
{
    __shared__ float lds[32 * 260];
    int tid = threadIdx.x;
    int i0 = blockIdx.x * 32;
    int b = i0 >> 12, s0 = i0 & (HWSP - 1);
    const float* qb = q + ((size_t)b * DDIM) * HWSP + s0;
    for (int j = tid; j < 32 * DDIM; j += 256) {
        int sl = j & 31, c = j >> 5;               // coalesced along spatial
        async_copy_b32(lds_off(&lds[sl * 260 + c]), qb + (size_t)c * HWSP + sl);
    }
    async_wait0();
    __syncthreads();

    int wave = tid >> 5, lane = tid & 31;
    float glAcc = 0.f, spAcc = 0.f;
    for (int rr = 0; rr < 4; ++rr) {
        int rl = wave + rr * 8;
        int i = i0 + rl;
        int g = gidx[i], g2 = sid2[i];
        float rn = rnorm[i];
        const float* kp = keys + (size_t)g * DDIM;
        const float* kn = keys + (size_t)g2 * DDIM;
        const float* qrow = &lds[rl * 260];
        float sg = 0.f, sp = 0.f, sn = 0.f, qv[8];
        for (int t = 0; t < 8; ++t) {
            int c = lane + 32 * t;
            float v = qrow[c] * rn;
            qv[t] = v;
            float dg = v - kp[c];
            sg += dg * dg;
            float dpe = dg + 1e-6f;        sp += dpe * dpe;
            float dne = v - kn[c] + 1e-6f; sn += dne * dne;
        }
        for (int off = 16; off > 0; off >>= 1) {
            sg += __shfl_xor(sg, off, 32);
            sp += __shfl_xor(sp, off, 32);
            sn += __shfl_xor(sn, off, 32);
        }
        float w = expf(rawscore[(size_t)i * MKEYS + g] - colmax[g]);
        for (int t = 0; t < 8; ++t) {
            int c = lane + 32 * t;
            atomicAdd(&qupd[(size_t)g * DDIM + c], w * qv[t]);
        }
        glAcc += sg;
        spAcc += fmaxf(sqrtf(sp) - sqrtf(sn) + 1.0f, 0.0f);
    }
    if (lane == 0) {
        atomicAdd(&accum[0], glAcc * (1.0f / ((float)NROWS * (float)DDIM)));
        atomicAdd(&accum[1], spAcc * (1.0f / (float)NROWS));
    }
}

// ---------------------------------------------------------------------------
// 6) GEMM2 (transposed): cm^T = keys^T @ sm^T, stored channel-major into
//    updated_query second half (coalesced stores). fp32 WMMA 16x16x4.
//    block = 128 thr (4 waves); tile = 64 channels x 16 spatial rows
//    sm tile staged via async Global->LDS in 16B chunks.
// ---------------------------------------------------------------------------
__global__ void k_gemm2(const float* __restrict__ sm,
                        const float* __restrict__ keys,
                        float* __restrict__ uq)
{
    __shared__ float ldsB[16 * 516];               // 16 rows(i) x 512 k, pad 4
    int tid = threadIdx.x;
    int i0 = blockIdx.x * 16;
    for (int j = tid; j < (16 * MKEYS) / 4; j += 128) {
        int f = j * 4;                             // 4-float chunk, never
        int r = f >> 9, k = f & (MKEYS - 1);       // crosses a 512-elem row
        async_copy_b128(lds_off(&ldsB[r * 516 + k]),
                        sm + (size_t)(i0 + r) * MKEYS + k);
    }
    async_wait0();
    __syncthreads();

    int wave = tid >> 5, lane = tid & 31;
    int c0 = blockIdx.y * 64 + wave * 16;
    int m = lane & 15, hi = lane >> 4;
    int b = i0 >> 12, s0 = i0 & (HWSP - 1);
    v8f acc = {};
    const float* kb = keys + c0 + m;               // A[m][k] = keys[k][c0+m]
    const float* bl = &ldsB[m * 516];              // B[k][n] = sm[i0+n][k]
    __builtin_prefetch(kb, 0, 1);
    for (int k0 = 0; k0 < MKEYS; k0 += 4) {
        int ka = k0 + 2 * hi;
        v2f a;
        a.x = kb[(size_t)ka * DDIM];
        a.y = kb[(size_t)(ka + 1) * DDIM];
        v2f bb = *(const v2f*)(bl + ka);
        acc = __builtin_amdgcn_wmma_f32_16x16x4_f32(
                  false, a, false, bb, (short)0, acc, false, false);
    }
    float* ob = uq + ((size_t)(b * (2 * DDIM) + DDIM + c0)) * HWSP + s0;
    for (int j = 0; j < 8; ++j) {
        int cl = j + 8 * hi;                       // channel within tile
        ob[(size_t)cl * HWSP + m] = acc[j];        // lanes 0..15 -> coalesced s
    }
}

// ---------------------------------------------------------------------------
// 7) updated_memory = l2norm(qupd + keys, axis=1)
// ---------------------------------------------------------------------------
__global__ void k_umem(const float* __restrict__ qupd,
                       const float* __restrict__ keys,
                       float* __restrict__ um)
{
    __shared__ float red[8];
    int mrow = blockIdx.x, c = threadIdx.x;
    float v = qupd[(size_t)mrow * DDIM + c] + keys[(size_t)mrow * DDIM + c];
    float ss = v * v;
    for (int off = 16; off > 0; off >>= 1) ss += __shfl_xor(ss, off, 32);
    int wave = c >> 5, lane = c & 31;
    if (lane == 0) red[wave] = ss;
    __syncthreads();
    float tot = 0.f;
    for (int j = 0; j < 8; ++j) tot += red[j];
    um[(size_t)mrow * DDIM + c] = v / fmaxf(sqrtf(tot), 1e-12f);
}

__global__ void k_final(const float* __restrict__ accum, float* __restrict__ losses)
{
    if (threadIdx.x == 0) { losses[0] = accum[0]; losses[1] = accum[1]; }
}

// ---------------------------------------------------------------------------
extern "C" void kernel_launch(void* const* d_in, const int* in_sizes, int n_in,
                              void* d_out, int out_size, void* d_ws, size_t ws_size,
                              hipStream_t stream)
{
    const float* query = (const float*)d_in[0];    // 32*256*64*64
    const float* keys  = (const float*)d_in[1];    // 512*256

    float* out = (float*)d_out;
    float* uq = out;                                // 67,108,864
    float* um = uq + (size_t)32 * 512 * HWSP;       //    131,072
    float* sq = um + (size_t)MKEYS * DDIM;          // 67,108,864 (raw scratch)
    float* sm = sq + (size_t)NROWS * MKEYS;         // 67,108,864
    float* losses = sm + (size_t)NROWS * MKEYS;     // [gather, spread]

    // workspace layout
    float* rnorm  = (float*)d_ws;                   // N
    int*   gidx   = (int*)(rnorm + NROWS);          // N
    int*   sid2   = gidx + NROWS;                   // N
    float* colmax = (float*)(sid2 + NROWS);         // M
    float* colsum = colmax + MKEYS;                 // M
    float* pmax   = colsum + MKEYS;                 // 64*M
    float* psum   = pmax + 64 * MKEYS;              // 64*M
    float* qupd   = psum + 64 * MKEYS;              // M*DDIM
    float* accum  = qupd + (size_t)MKEYS * DDIM;    // 2

    // zero the accumulated buffers each call (graph-capturable)
    hipMemsetAsync(qupd, 0, ((size_t)MKEYS * DDIM + 2) * sizeof(float), stream);

    k_norm<<<NROWS / 256, 256, 0, stream>>>(query, rnorm, uq);
    k_gemm1<<<dim3(NROWS / 16, MKEYS / 64), 128, 0, stream>>>(query, keys, rnorm, sq);
    k_rowsoftmax<<<NROWS / 8, 256, 0, stream>>>(sq, sm, gidx, sid2);
    k_colmax_part<<<dim3(MKEYS / 256, 64), 256, 0, stream>>>(sq, pmax);
    k_colmax_red<<<MKEYS / 256, 256, 0, stream>>>(pmax, colmax);
    k_colsum_part<<<dim3(MKEYS / 256, 64), 256, 0, stream>>>(sq, colmax, psum);
    k_colsum_red<<<MKEYS / 256, 256, 0, stream>>>(psum, colsum);
    k_rowops<<<NROWS / 32, 256, 0, stream>>>(query, keys, rnorm, sq, colmax,
                                             gidx, sid2, qupd, accum);
    k_sq_transform<<<(NROWS * (MKEYS / 256)), 256, 0, stream>>>(sq, colmax, colsum);
    k_gemm2<<<dim3(NROWS / 16, DDIM / 64), 128, 0, stream>>>(sm, keys, uq);
    k_umem<<<MKEYS, 256, 0, stream>>>(qupd, keys, um);
    k_final<<<1, 32, 0, stream>>>(accum, losses);
}